// GAT_12232066859439
// MI455X (gfx1250) — compile-verified
//
#include <hip/hip_runtime.h>
#include <math.h>

// ---------------------------------------------------------------------------
// GAT (3 layers, N=1024) + tiny MLP head, for MI455X / gfx1250 (wave32, WMMA)
// ---------------------------------------------------------------------------
//  - h = x@W and att@h GEMMs use V_WMMA_F32_16X16X4_F32 (fp32 in/out).
//  - e2 collapses to a (N/f) x f table E2 (see torch-view algebra), so the
//    attention-logit pass is O(N^2) cheap VALU work.
//  - Head layer-1 streams 402 MB of l1w: float4 coalesced + global_prefetch.
// ---------------------------------------------------------------------------

#define GN      1024
#define GALPHA  0.2f
#define GNEG    (-9.0e15f)

typedef float v2f __attribute__((ext_vector_type(2)));
typedef float v8f __attribute__((ext_vector_type(8)));

#if defined(__has_builtin)
# if __has_builtin(__builtin_amdgcn_wmma_f32_16x16x4_f32)
#  define HAVE_WMMA_F32 1
# endif
#endif
#ifndef HAVE_WMMA_F32
# define HAVE_WMMA_F32 0
#endif

// ---------------------------------------------------------------------------
// C[M,Nc] = A[M,K] @ B[K,Nc]; one wave32 per 16x16 tile; fp32 WMMA, K-step 4.
// mode 0: C = z            mode 1: C = relu(relu(z) + bias[col])
// blockDim = (32,4): 4 tiles per block. Guards are wave-uniform so EXEC is
// all-ones around every WMMA (ISA requirement).
// ---------------------------------------------------------------------------
__global__ void wmma_gemm_f32(const float* __restrict__ A,
                              const float* __restrict__ B,
                              float* __restrict__ C,
                              int M, int K, int Nc,
                              const float* __restrict__ bias,
                              int mode, int ntiles)
{
    int tile = blockIdx.x * blockDim.y + threadIdx.y;   // wave-uniform
    if (tile >= ntiles) return;
    int ntN = Nc >> 4;
    int tm  = (tile / ntN) << 4;
    int tn  = (tile % ntN) << 4;

    int lane = threadIdx.x;        // 0..31
    int half = lane >> 4;          // 0: K=0,1   1: K=2,3
    int l16  = lane & 15;

    v8f acc = {};
    const float* Arow = A + (size_t)(tm + l16) * K;
    for (int k = 0; k < K; k += 4) {
        int ka = k + half * 2;
        float2 av = *(const float2*)(Arow + ka);            // A[m, ka..ka+1]
        v2f a; a.x = av.x; a.y = av.y;
        v2f b;
        b.x = B[(size_t)ka       * Nc + tn + l16];          // B[ka,   n]
        b.y = B[(size_t)(ka + 1) * Nc + tn + l16];          // B[ka+1, n]
#if HAVE_WMMA_F32
        acc = __builtin_amdgcn_wmma_f32_16x16x4_f32(
                  /*neg_a=*/false, a, /*neg_b=*/false, b,
                  /*c_mod=*/(short)0, acc,
                  /*reuse_a=*/false, /*reuse_b=*/false);
#else
        // Scalar emulation with identical register layout (compile hedge).
        for (int kk = 0; kk < 4; ++kk) {
            float bk = B[(size_t)(k + kk) * Nc + tn + l16];
            #pragma unroll
            for (int v = 0; v < 8; ++v)
                acc[v] += A[(size_t)(tm + v + half * 8) * K + k + kk] * bk;
        }
#endif
    }
    // C/D layout: VGPR v -> row tm + v (lanes 0-15) / tm + v + 8 (lanes 16-31)
    #pragma unroll
    for (int v = 0; v < 8; ++v) {
        int r = tm + v + half * 8;
        float z = acc[v];
        if (mode == 1) {
            z = fmaxf(z, 0.0f) + bias[tn + l16];
            z = fmaxf(z, 0.0f);
        }
        C[(size_t)r * Nc + tn + l16] = z;
    }
}

// ---------------------------------------------------------------------------
// e1[i] = h[i,:] . a[:f]            (i in [0,N))
// E2[g*f+m] = sum_p a[f+p] * h[(N/f)*p + g, m]   (g in [0,N/f), m in [0,f))
// 2048 threads total.
// ---------------------------------------------------------------------------
__global__ void e1_e2_kernel(const float* __restrict__ h,
                             const float* __restrict__ a,
                             float* __restrict__ e1,
                             float* __restrict__ E2, int f)
{
    int tid = blockIdx.x * blockDim.x + threadIdx.x;
    if (tid < GN) {
        const float* hr = h + (size_t)tid * f;
        float s = 0.0f;
        for (int m = 0; m < f; ++m) s += hr[m] * a[m];
        e1[tid] = s;
    } else if (tid < 2 * GN) {
        int idx = tid - GN;
        int g = idx / f, m = idx - g * f;
        int stride = GN / f;
        float s = 0.0f;
        for (int p = 0; p < f; ++p)
            s += a[f + p] * h[(size_t)(stride * p + g) * f + m];
        E2[idx] = s;
    }
}

// ---------------------------------------------------------------------------
// One wave32 per row r:  att[r,:] = softmax_c( mask ? leaky(e1[c]+E2[r//f,c%f])
//                                                   : NEG )
// Each lane owns 32 columns; shfl_xor reductions (wave32).
// ---------------------------------------------------------------------------
__global__ void att_rows_kernel(const float* __restrict__ e1,
                                const float* __restrict__ E2,
                                const int* __restrict__ adj,
                                float* __restrict__ att, int f)
{
    int r    = blockIdx.x * (blockDim.x >> 5) + (threadIdx.x >> 5);
    int lane = threadIdx.x & 31;
    const float* E2row = E2 + (size_t)(r / f) * f;
    const int*   arow  = adj + (size_t)r * GN;
    float*       orow  = att + (size_t)r * GN;
    int fm = f - 1;                     // f is a power of two

    float ev[32];
    float mx = -INFINITY;
    #pragma unroll
    for (int i = 0; i < 32; ++i) {
        int c = lane + i * 32;
        float e = e1[c] + E2row[c & fm];
        e = (e > 0.0f) ? e : GALPHA * e;          // leaky_relu before mask
        e = (arow[c] > 0) ? e : GNEG;
        ev[i] = e;
        mx = fmaxf(mx, e);
    }
    #pragma unroll
    for (int o = 16; o > 0; o >>= 1) mx = fmaxf(mx, __shfl_xor(mx, o, 32));

    float sum = 0.0f;
    #pragma unroll
    for (int i = 0; i < 32; ++i) { ev[i] = __expf(ev[i] - mx); sum += ev[i]; }
    #pragma unroll
    for (int o = 16; o > 0; o >>= 1) sum += __shfl_xor(sum, o, 32);

    float inv = 1.0f / sum;
    #pragma unroll
    for (int i = 0; i < 32; ++i) orow[lane + i * 32] = ev[i] * inv;
}

// ---------------------------------------------------------------------------
// Head layer 1: t1[bi] = relu( l1w[bi, 0:65536] . xf + l1b[bi] ), bi < 1536.
// One block per output; float4 streaming (dominant 402 MB read) + prefetch.
// ---------------------------------------------------------------------------
__global__ void head_l1_kernel(const float* __restrict__ w,
                               const float* __restrict__ b,
                               const float* __restrict__ xf,
                               float* __restrict__ t1)
{
    const int D4 = 65536 / 4;          // 16384 float4 per output row
    int bi = blockIdx.x;
    const float4* w4 = (const float4*)(w + (size_t)bi * 65536);
    const float4* x4 = (const float4*)xf;

    float s = 0.0f;
    for (int i = threadIdx.x; i < D4; i += 256) {
        if (i + 2048 < D4) __builtin_prefetch(&w4[i + 2048], 0, 0);
        float4 a = w4[i], c = x4[i];
        s += a.x * c.x + a.y * c.y + a.z * c.z + a.w * c.w;
    }
    __shared__ float red[256];
    red[threadIdx.x] = s;
    __syncthreads();
    for (int o = 128; o > 0; o >>= 1) {
        if (threadIdx.x < o) red[threadIdx.x] += red[threadIdx.x + o];
        __syncthreads();
    }
    if (threadIdx.x == 0) t1[bi] = fmaxf(red[0] + b[bi], 0.0f);
}

// Head layer 2: t2[h*128+o] = sigmoid( l2w[h,o,:] . t1[h,:] + l2b[h,o] )
__global__ void head_l2_kernel(const float* __restrict__ w,
                               const float* __restrict__ b,
                               const float* __restrict__ t1,
                               float* __restrict__ t2)
{
    int h = blockIdx.x;                // 0..5
    int o = threadIdx.x;               // 0..127
    const float* wr = w + ((size_t)h * 128 + o) * 256;
    const float* tr = t1 + (size_t)h * 256;
    float s = 0.0f;
    for (int t = 0; t < 256; ++t) s += wr[t] * tr[t];
    s += b[h * 128 + o];
    t2[h * 128 + o] = 1.0f / (1.0f + __expf(-s));
}

// Head layer 3: out[h] = l3w[h,0,:] . t2[h,:] + l3b[h]
__global__ void head_l3_kernel(const float* __restrict__ w,
                               const float* __restrict__ b,
                               const float* __restrict__ t2,
                               float* __restrict__ out)
{
    int h = blockIdx.x;                // 0..5
    int t = threadIdx.x;               // 0..127
    __shared__ float red[128];
    red[t] = w[(size_t)h * 128 + t] * t2[(size_t)h * 128 + t];
    __syncthreads();
    for (int o = 64; o > 0; o >>= 1) {
        if (t < o) red[t] += red[t + o];
        __syncthreads();
    }
    if (t == 0) out[h] = red[0] + b[h];
}

// ---------------------------------------------------------------------------
// Host-side orchestration
// ---------------------------------------------------------------------------
static void run_gat_layer(const float* in, int inF,
                          const float* W, int f,
                          const float* a, const float* b,
                          const int* adj,
                          float* att, float* hA, float* out,
                          float* e1v, float* E2v, hipStream_t stream)
{
    dim3 blk(32, 4);
    int tiles1 = (GN / 16) * (f / 16);
    wmma_gemm_f32<<<dim3((tiles1 + 3) / 4), blk, 0, stream>>>(
        in, W, hA, GN, inF, f, (const float*)nullptr, 0, tiles1);

    e1_e2_kernel<<<dim3(8), dim3(256), 0, stream>>>(hA, a, e1v, E2v, f);

    att_rows_kernel<<<dim3(GN / 8), dim3(256), 0, stream>>>(e1v, E2v, adj, att, f);

    int tiles2 = (GN / 16) * (f / 16);
    wmma_gemm_f32<<<dim3((tiles2 + 3) / 4), blk, 0, stream>>>(
        att, hA, out, GN, GN, f, b, 1, tiles2);
}

extern "C" void kernel_launch(void* const* d_in, const int* in_sizes, int n_in,
                              void* d_out, int out_size, void* d_ws, size_t ws_size,
                              hipStream_t stream)
{
    (void)in_sizes; (void)n_in; (void)out_size; (void)ws_size;

    const float* x   = (const float*)d_in[0];
    const int*   adj = (const int*)  d_in[1];
    const float* W1  = (const float*)d_in[2];
    const float* a1  = (const float*)d_in[3];
    const float* b1  = (const float*)d_in[4];
    const float* W2  = (const float*)d_in[5];
    const float* a2  = (const float*)d_in[6];
    const float* b2  = (const float*)d_in[7];
    const float* W3  = (const float*)d_in[8];
    const float* a3  = (const float*)d_in[9];
    const float* b3  = (const float*)d_in[10];
    const float* l1w = (const float*)d_in[11];
    const float* l1b = (const float*)d_in[12];
    const float* l2w = (const float*)d_in[13];
    const float* l2b = (const float*)d_in[14];
    const float* l3w = (const float*)d_in[15];
    const float* l3b = (const float*)d_in[16];

    float* ws  = (float*)d_ws;
    float* att = ws;                       // 1024*1024
    float* hA  = att + 1024 * 1024;        // 1024*128 (scratch h)
    float* hB  = hA  + 1024 * 128;         // 1024*128 (layer outputs)
    float* e1v = hB  + 1024 * 128;         // 1024
    float* E2v = e1v + 1024;               // 1024
    float* t1  = E2v + 1024;               // 6*256
    float* t2  = t1  + 6 * 256;            // 6*128

    // Layer 1: x[1024,512] -> h1 = hB[1024,128]
    run_gat_layer(x,  512, W1, 128, a1, b1, adj, att, hA, hB, e1v, E2v, stream);
    // Layer 2: h1[1024,128] -> h2 = hB[1024,64]
    run_gat_layer(hB, 128, W2,  64, a2, b2, adj, att, hA, hB, e1v, E2v, stream);
    // Layer 3: h2[1024,64] -> h3 = hB[1024,64]  (== xf flattened)
    run_gat_layer(hB,  64, W3,  64, a3, b3, adj, att, hA, hB, e1v, E2v, stream);

    // Head (l1 streams 402 MB of weights — the roofline-dominant step)
    head_l1_kernel<<<dim3(6 * 256), dim3(256), 0, stream>>>(l1w, l1b, hB, t1);
    head_l2_kernel<<<dim3(6),       dim3(128), 0, stream>>>(l2w, l2b, t1, t2);
    head_l3_kernel<<<dim3(6),       dim3(128), 0, stream>>>(l3w, l3b, t2, (float*)d_out);
}